// MultiHeadSelfAttention_49306224558176
// MI455X (gfx1250) — compile-verified
//
#include <hip/hip_runtime.h>

typedef float v2f __attribute__((ext_vector_type(2)));
typedef float v8f __attribute__((ext_vector_type(8)));

#define EMBED   768
#define THREE_E 2304
#define NHEAD   12
#define HDIM    64
#define SEQ     1024
#define BATCH   8
#define ROWS    (BATCH * SEQ)   // 8192

// ---------------------------------------------------------------------------
// fp32 GEMM via V_WMMA_F32_16X16X4_F32.
// C[M,N] = A[M,K] @ B[K,N] (+ bias). 256 threads = 8 waves.
// Block tile 128(M) x 128(N); each wave computes a 16x128 strip.
// LDS is k-minor (stride 20 floats = 80B): fragment pair loads are single
// aligned ds_load_b64 into even VGPR pairs, bank-conflict-free.
// ---------------------------------------------------------------------------
__global__ __launch_bounds__(256) void gemm_f32_wmma(
    const float* __restrict__ A, const float* __restrict__ B,
    const float* __restrict__ bias, float* __restrict__ C,
    int M, int N, int K)
{
    __shared__ float As[128][20];   // [m][k] , 80B row stride
    __shared__ float Bs[128][20];   // [n][k] , transposed at staging

    const int tid   = threadIdx.x;
    const int wave  = tid >> 5;
    const int lane  = tid & 31;
    const int l15   = lane & 15;
    const int kp    = (lane >> 4) * 2;   // lanes 0-15: K0/K1, 16-31: K2/K3
    const int tileM = blockIdx.y * 128;
    const int tileN = blockIdx.x * 128;

    v8f acc[8] = {};

    for (int k0 = 0; k0 < K; k0 += 16) {
        // Stage A 128x16 -> As[m][k]  (512 float4, 2/thread, 16B-aligned LDS)
        #pragma unroll
        for (int i = 0; i < 2; ++i) {
            int idx = tid + i * 256;          // 0..511
            int m   = idx >> 2;               // 0..127
            int c4  = (idx & 3) * 4;          // 0,4,8,12
            const float* src = A + (size_t)(tileM + m) * K + k0 + c4;
            float4 v = *(const float4*)src;
            *(float4*)&As[m][c4] = v;
            if (k0 + 16 < K) __builtin_prefetch(src + 16, 0, 0);
        }
        // Stage B 16x128 -> Bs[n][k]  (coalesced global read, transposed store)
        #pragma unroll
        for (int i = 0; i < 2; ++i) {
            int idx = tid + i * 256;          // 0..511
            int kr  = idx >> 5;               // 0..15
            int c4  = (idx & 31) * 4;         // 0..124
            const float* src = B + (size_t)(k0 + kr) * N + tileN + c4;
            float4 v = *(const float4*)src;
            Bs[c4 + 0][kr] = v.x;
            Bs[c4 + 1][kr] = v.y;
            Bs[c4 + 2][kr] = v.z;
            Bs[c4 + 3][kr] = v.w;
            if (k0 + 16 < K) __builtin_prefetch(src + (size_t)16 * N, 0, 0);
        }
        __syncthreads();

        const int am = wave * 16 + l15;
        #pragma unroll
        for (int kk = 0; kk < 16; kk += 4) {
            v2f a = *(const v2f*)&As[am][kk + kp];          // ds_load_b64
            #pragma unroll
            for (int j = 0; j < 8; ++j) {
                v2f b = *(const v2f*)&Bs[j * 16 + l15][kk + kp];
                acc[j] = __builtin_amdgcn_wmma_f32_16x16x4_f32(
                    false, a, false, b, (short)0, acc[j], false, false);
            }
        }
        __syncthreads();
    }

    // C tile write: VGPR r holds (M = r + 8*(lane>=16), N = l15)
    #pragma unroll
    for (int j = 0; j < 8; ++j) {
        int ncol = tileN + j * 16 + l15;
        float bv = bias ? bias[ncol] : 0.0f;
        #pragma unroll
        for (int r = 0; r < 8; ++r) {
            int mrow = tileM + wave * 16 + r + (lane >> 4) * 8;
            C[(size_t)mrow * N + ncol] = acc[j][r] + bv;
        }
    }
}

// ---------------------------------------------------------------------------
// Fused attention: S = QK^T*scale -> softmax -> write attn -> O = P@V.
// One block = one (b, h, 16-row tile of queries). 256 threads = 8 waves.
// ---------------------------------------------------------------------------
__global__ __launch_bounds__(256) void attn_f32_wmma(
    const float* __restrict__ QKV,   // [8192, 2304] rows of (Q|K|V)
    float* __restrict__ attn,        // [B, H, N, N]
    float* __restrict__ Ows)         // [8192, 768] as [B, N, H, D]
{
    __shared__ float S[16][1025];        // scores / probabilities (padded)
    __shared__ float Qs[16][68];         // Q tile 16x64 (272B rows: 16B aligned)
    __shared__ float Ks[8][16][68];      // per-wave K tile slabs
    __shared__ float red[16][16];
    __shared__ float rowmax[16];
    __shared__ float rowinv[16];

    const int tid  = threadIdx.x;
    const int wave = tid >> 5;
    const int lane = tid & 31;
    const int l15  = lane & 15;
    const int kp   = (lane >> 4) * 2;

    const int bx = blockIdx.x;
    const int qt = bx & 63;                  // query row-tile within sequence
    const int h  = (bx >> 6) % NHEAD;
    const int b  = bx / (64 * NHEAD);

    const size_t rstride = THREE_E;
    const size_t base    = (size_t)b * SEQ * rstride;
    const int qcol = h * HDIM;
    const int kcol = EMBED + h * HDIM;
    const int vcol = 2 * EMBED + h * HDIM;
    const int q0   = qt * 16;

    // ---- stage Q tile 16x64 (coalesced float4) ----
    {
        int r  = tid >> 4;
        int c4 = (tid & 15) * 4;
        float4 v = *(const float4*)(QKV + base + (size_t)(q0 + r) * rstride + qcol + c4);
        *(float4*)&Qs[r][c4] = v;
    }
    __syncthreads();

    // ---- scores: wave w covers columns [w*128, w*128+128) in 8 tiles ----
    for (int j = 0; j < 8; ++j) {
        const int n0 = (wave * 8 + j) * 16;
        // stage K tile 16x64 into this wave's slab (coalesced)
        #pragma unroll
        for (int i = 0; i < 8; ++i) {
            int idx = lane + i * 32;         // 0..255
            int r   = idx >> 4;
            int c4  = (idx & 15) * 4;
            float4 v = *(const float4*)(QKV + base + (size_t)(n0 + r) * rstride + kcol + c4);
            *(float4*)&Ks[wave][r][c4] = v;
        }
        v8f c = {};
        #pragma unroll
        for (int d = 0; d < HDIM; d += 4) {
            v2f a  = *(const v2f*)&Qs[l15][d + kp];          // ds_load_b64
            v2f kb = *(const v2f*)&Ks[wave][l15][d + kp];    // B[k][n] = K[n][d+k]
            c = __builtin_amdgcn_wmma_f32_16x16x4_f32(
                false, a, false, kb, (short)0, c, false, false);
        }
        #pragma unroll
        for (int r = 0; r < 8; ++r)
            S[r + (lane >> 4) * 8][n0 + l15] = c[r];
    }
    __syncthreads();

    // ---- softmax over 1024 columns, 16 threads per row ----
    const float scale = 0.125f;  // 1/sqrt(64)
    {
        int r = tid >> 4, c16 = tid & 15;
        float mx = -3.402823466e38f;
        for (int c = c16; c < SEQ; c += 16) mx = fmaxf(mx, S[r][c]);
        red[r][c16] = mx;
    }
    __syncthreads();
    if ((tid & 15) == 0) {
        int r = tid >> 4;
        float mx = red[r][0];
        #pragma unroll
        for (int i = 1; i < 16; ++i) mx = fmaxf(mx, red[r][i]);
        rowmax[r] = mx;
    }
    __syncthreads();
    {
        int r = tid >> 4, c16 = tid & 15;
        float mx = rowmax[r];
        float s = 0.0f;
        for (int c = c16; c < SEQ; c += 16) {
            float e = __expf((S[r][c] - mx) * scale);
            S[r][c] = e;
            s += e;
        }
        red[r][c16] = s;
    }
    __syncthreads();
    if ((tid & 15) == 0) {
        int r = tid >> 4;
        float s = 0.0f;
        #pragma unroll
        for (int i = 0; i < 16; ++i) s += red[r][i];
        rowinv[r] = 1.0f / s;
    }
    __syncthreads();

    // ---- normalize in LDS + write attn (coalesced 4KB rows) ----
    float* arow = attn + ((size_t)(b * NHEAD + h) * SEQ + q0) * SEQ;
    for (int i = 0; i < 64; ++i) {
        int idx = tid + i * 256;             // 0..16383
        int r = idx >> 10, c = idx & 1023;
        float p = S[r][c] * rowinv[r];
        S[r][c] = p;
        arow[(size_t)r * SEQ + c] = p;
    }
    __syncthreads();

    // ---- O = P @ V : waves 0..3 each own one 16-wide d-tile ----
    if (wave < 4) {
        const int d0 = wave * 16;
        const float* Vb = QKV + base + vcol + d0 + l15;
        v8f c = {};
        for (int nk = 0; nk < SEQ; nk += 4) {
            v2f a, vb;
            a.x = S[l15][nk + kp + 0];       // A = P (M x k); 1025 stride -> b32s
            a.y = S[l15][nk + kp + 1];
            const float* vp = Vb + (size_t)(nk + kp) * rstride;
            vb.x = vp[0];                    // B[k][n] = V[nk+k][d]
            vb.y = vp[rstride];
            c = __builtin_amdgcn_wmma_f32_16x16x4_f32(
                false, a, false, vb, (short)0, c, false, false);
        }
        float* orow = Ows + (size_t)(b * SEQ + q0) * EMBED + h * HDIM + d0;
        #pragma unroll
        for (int r = 0; r < 8; ++r) {
            int m = r + (lane >> 4) * 8;
            orow[(size_t)m * EMBED + l15] = c[r];
        }
    }
}

// ---------------------------------------------------------------------------
extern "C" void kernel_launch(void* const* d_in, const int* in_sizes, int n_in,
                              void* d_out, int out_size, void* d_ws, size_t ws_size,
                              hipStream_t stream) {
    const float* x      = (const float*)d_in[0];   // [8,1024,768]
    const float* w_qkv  = (const float*)d_in[1];   // [768,2304]
    const float* w_proj = (const float*)d_in[2];   // [768,768]
    const float* b_proj = (const float*)d_in[3];   // [768]

    float* out  = (float*)d_out;                               // [8,1024,768]
    float* attn = out + (size_t)ROWS * EMBED;                  // [8,12,1024,1024]

    float* qkv = (float*)d_ws;                                 // [8192,2304]
    float* ows = qkv + (size_t)ROWS * THREE_E;                 // [8192,768]

    dim3 blk(256);

    // 1) QKV projection: [8192,768] @ [768,2304]
    gemm_f32_wmma<<<dim3(THREE_E / 128, ROWS / 128), blk, 0, stream>>>(
        x, w_qkv, nullptr, qkv, ROWS, THREE_E, EMBED);

    // 2) fused attention (writes attn + O workspace)
    attn_f32_wmma<<<dim3(BATCH * NHEAD * (SEQ / 16)), blk, 0, stream>>>(
        qkv, attn, ows);

    // 3) output projection: [8192,768] @ [768,768] + bias
    gemm_f32_wmma<<<dim3(EMBED / 128, ROWS / 128), blk, 0, stream>>>(
        ows, w_proj, b_proj, out, ROWS, EMBED, EMBED);
}